// GILR_LSTM_32865089749626
// MI455X (gfx1250) — compile-verified
//
#include <hip/hip_runtime.h>
#include <stdint.h>

// Problem geometry (from the reference)
#define NSTEPS 65536
#define NDIMS  512

// Tiling: one block = 256 channels x 64 time steps, LDS tile = 2*64*256*4B = 128KB
#define CHUNK  64
#define NCHUNK (NSTEPS / CHUNK)   // 1024 time chunks
#define CPB    256                // channels per block
#define NGRP   (NDIMS / CPB)      // 2 channel groups
#define TPB    256                // 8 wave32s
#define V4_PER_ROW (CPB / 4)      // 64 float4 per LDS row
#define V4_ITERS  ((CHUNK * CPB / 4) / TPB)  // 16 float4 copies per thread

#define GLOBAL_AS __attribute__((address_space(1)))
#define LDS_AS    __attribute__((address_space(3)))

typedef int v4i __attribute__((vector_size(16)));   // matches builtin param type

#if __has_builtin(__builtin_amdgcn_global_load_async_to_lds_b128)
#define HAVE_ASYNC_LD 1
#else
#define HAVE_ASYNC_LD 0
#endif
#if __has_builtin(__builtin_amdgcn_global_store_async_from_lds_b128)
#define HAVE_ASYNC_ST 1
#else
#define HAVE_ASYNC_ST 0
#endif

// Generic LDS pointer on amdgcn = {aperture_hi32, lds_offset32}: low 32 bits are the
// LDS byte offset, which is exactly what an AS(3) pointer holds.
__device__ __forceinline__ LDS_AS v4i* to_lds_v4(const void* p) {
  return (LDS_AS v4i*)(uint32_t)(uintptr_t)p;
}
__device__ __forceinline__ GLOBAL_AS v4i* to_global_v4(const void* p) {
  return (GLOBAL_AS v4i*)(uintptr_t)p;
}

__device__ __forceinline__ void wait_async0() {
#if __has_builtin(__builtin_amdgcn_s_wait_asynccnt)
  __builtin_amdgcn_s_wait_asynccnt(0);
#else
  asm volatile("s_wait_asynccnt 0" ::: "memory");
#endif
}

__global__ void zero_flags_kernel(int* __restrict__ flags, int n) {
  int i = blockIdx.x * blockDim.x + threadIdx.x;
  if (i < n) flags[i] = 0;
}

// Single-pass chunked linear-recurrence scan with decoupled lookback.
// h[t] = a[t]*h[t-1] + b[t];  pair combine: (a1,b1) then (a2,b2) -> (a1*a2, a2*b1 + b2)
__global__ __launch_bounds__(TPB, 1) void gilr_scan_kernel(
    const float* __restrict__ x, float* __restrict__ out,
    float* __restrict__ aggA, float* __restrict__ aggB,
    float* __restrict__ incA, float* __restrict__ incB,
    int*   __restrict__ flags)
{
  __shared__ float sG[CHUNK][CPB];   // gate_raw -> a -> h
  __shared__ float sI[CHUNK][CPB];   // impulse_raw -> b
  __shared__ int   sFlag;

  const int j   = blockIdx.x;        // time chunk index
  const int g   = blockIdx.y;        // channel group
  const int tid = threadIdx.x;
  const int c   = g * CPB + tid;     // this thread's channel
  const size_t row0 = (size_t)j * CHUNK;

  // ---------------- Phase 1: stage raw tiles into LDS (async data mover path) ----
  #pragma unroll
  for (int k = 0; k < V4_ITERS; ++k) {
    int f  = tid + k * TPB;
    int t  = f / V4_PER_ROW;
    int c4 = (f % V4_PER_ROW) * 4;
    const float* gp = x + (row0 + t) * NDIMS + g * CPB + c4;
    const float* ip = gp + (size_t)NSTEPS * NDIMS;
#if HAVE_ASYNC_LD
    __builtin_amdgcn_global_load_async_to_lds_b128(to_global_v4(gp), to_lds_v4(&sG[t][c4]), 0, 0);
    __builtin_amdgcn_global_load_async_to_lds_b128(to_global_v4(ip), to_lds_v4(&sI[t][c4]), 0, 0);
#else
    *(float4*)&sG[t][c4] = *(const float4*)gp;
    *(float4*)&sI[t][c4] = *(const float4*)ip;
#endif
  }
#if HAVE_ASYNC_LD
  wait_async0();
#endif
  __syncthreads();

  // ---------------- Phase 2: elementwise a,b + chunk-local aggregate -------------
  float A = 1.0f, B = 0.0f;          // (prod a, local scan end) with h0 = 0
  #pragma unroll 8
  for (int t = 0; t < CHUNK; ++t) {
    float gr = sG[t][tid];
    float ir = sI[t][tid];
    float a  = 1.0f / (1.0f + __expf(-gr));    // sigmoid
    float b  = (1.0f - a) * fmaxf(ir, 0.0f);   // (1-gate)*relu(impulse)
    sG[t][tid] = a;
    sI[t][tid] = b;
    A *= a;
    B = fmaf(a, B, b);
  }

  // ---------------- Phase 3: publish + decoupled lookback ------------------------
  const size_t pj = (size_t)j * NDIMS + c;
  float eA = 1.0f, eB = 0.0f;        // exclusive prefix pair for this chunk
  if (j == 0) {
    incA[pj] = A; incB[pj] = B;
    __threadfence();
    __syncthreads();
    if (tid == 0)
      __hip_atomic_store(&flags[j * NGRP + g], 2, __ATOMIC_RELEASE, __HIP_MEMORY_SCOPE_AGENT);
  } else {
    aggA[pj] = A; aggB[pj] = B;
    __threadfence();
    __syncthreads();
    if (tid == 0)
      __hip_atomic_store(&flags[j * NGRP + g], 1, __ATOMIC_RELEASE, __HIP_MEMORY_SCOPE_AGENT);

    int k = j - 1;
    for (;;) {
      if (tid == 0) {
        int f;
        while ((f = __hip_atomic_load(&flags[k * NGRP + g], __ATOMIC_ACQUIRE,
                                      __HIP_MEMORY_SCOPE_AGENT)) == 0)
          __builtin_amdgcn_s_sleep(1);
        sFlag = f;
      }
      __syncthreads();
      int f = sFlag;
      // make the published pairs visible to every lane in the block
      __builtin_amdgcn_fence(__ATOMIC_ACQUIRE, "agent");
      size_t pk = (size_t)k * NDIMS + c;
      float pA = (f == 2) ? incA[pk] : aggA[pk];
      float pB = (f == 2) ? incB[pk] : aggB[pk];
      float nA = pA * eA;            // prepend chunk k: (pA,pB) o (eA,eB)
      float nB = fmaf(eA, pB, eB);
      eA = nA; eB = nB;
      __syncthreads();               // protect sFlag reuse
      if (f == 2) break;             // f is block-uniform (came from sFlag)
      --k;
    }
    // inclusive = exclusive o local : (eA*A, A*eB + B)
    incA[pj] = eA * A;
    incB[pj] = fmaf(A, eB, B);
    __threadfence();
    __syncthreads();
    if (tid == 0)
      __hip_atomic_store(&flags[j * NGRP + g], 2, __ATOMIC_RELEASE, __HIP_MEMORY_SCOPE_AGENT);
  }

  // ---------------- Phase 4: replay scan with correct h0, write out --------------
  float h = eB;                      // state entering this chunk (0 for j==0)
  #pragma unroll 8
  for (int t = 0; t < CHUNK; ++t) {
    float a = sG[t][tid];
    float b = sI[t][tid];
    h = fmaf(a, h, b);
    sG[t][tid] = h;                  // stage result for cooperative vector stores
  }
  __syncthreads();

  #pragma unroll
  for (int k = 0; k < V4_ITERS; ++k) {
    int f  = tid + k * TPB;
    int t  = f / V4_PER_ROW;
    int c4 = (f % V4_PER_ROW) * 4;
    float* op = out + (row0 + t) * NDIMS + g * CPB + c4;
#if HAVE_ASYNC_ST
    __builtin_amdgcn_global_store_async_from_lds_b128(to_global_v4(op), to_lds_v4(&sG[t][c4]), 0, 0);
#else
    *(float4*)op = *(const float4*)&sG[t][c4];
#endif
  }
#if HAVE_ASYNC_ST
  wait_async0();                     // drain ASYNCcnt before wave exit
#endif
}

extern "C" void kernel_launch(void* const* d_in, const int* in_sizes, int n_in,
                              void* d_out, int out_size, void* d_ws, size_t ws_size,
                              hipStream_t stream) {
  const float* x = (const float*)d_in[0];
  float* out = (float*)d_out;

  // Workspace layout: 4 pair arrays [NCHUNK][NDIMS] f32 (8 MB) + flags [NCHUNK*NGRP]
  const size_t pairElems = (size_t)NCHUNK * NDIMS;
  float* aggA = (float*)d_ws;
  float* aggB = aggA + pairElems;
  float* incA = aggB + pairElems;
  float* incB = incA + pairElems;
  int*   flags = (int*)(incB + pairElems);
  const int nflags = NCHUNK * NGRP;

  // Flags must be re-zeroed every call (harness does not re-poison between replays)
  zero_flags_kernel<<<(nflags + 255) / 256, 256, 0, stream>>>(flags, nflags);

  dim3 grid(NCHUNK, NGRP, 1);
  gilr_scan_kernel<<<grid, TPB, 0, stream>>>(x, out, aggA, aggB, incA, incB, flags);
}